// GAT_7799660610001
// MI455X (gfx1250) — compile-verified
//
#include <hip/hip_runtime.h>
#include <cstdint>

// ---------------------------------------------------------------------------
// 2-layer GAT (PyG GATConv math) for MI455X / gfx1250, wave32.
// Dense node GEMMs use V_WMMA_F32_16X16X4_F32; each wave owns a 16x(16*NT)
// output strip (A fragment loaded once per k-step, reused across NT WMMAs),
// and W is staged in LDS pair-interleaved so each B fragment is one
// ds_load_b64 instead of two strided global b32 loads.
// Segment softmax uses monotone-int atomicMax + float atomicAdd; edge
// gathers use float4 (global_load_b128).
// ---------------------------------------------------------------------------

typedef __attribute__((ext_vector_type(2))) float v2f;
typedef __attribute__((ext_vector_type(8))) float v8f;

#define NEG_SLOPE 0.2f

// monotone mapping float -> int so signed atomicMax gives float max ordering
__device__ __forceinline__ int enc_f32(float f) {
  int i = __float_as_int(f);
  return i >= 0 ? i : (i ^ 0x7fffffff);
}
__device__ __forceinline__ float dec_f32(int i) {
  return __int_as_float(i >= 0 ? i : (i ^ 0x7fffffff));
}

__device__ __forceinline__ float lrelu(float x) {
  return x > 0.0f ? x : NEG_SLOPE * x;
}

// ---------------- fill kernels (re-init workspace every call) --------------
__global__ void fill_f32(float* p, float v, int n) {
  int i = blockIdx.x * blockDim.x + threadIdx.x;
  if (i < n) p[i] = v;
}
__global__ void fill_i32(int* p, int v, int n) {
  int i = blockIdx.x * blockDim.x + threadIdx.x;
  if (i < n) p[i] = v;
}

// ---------------- fp32 WMMA GEMM:  Y[N,Fout] = X[N,K] * W[K,Fout] ----------
// One wave computes a 16 x (16*NT) strip with v_wmma_f32_16x16x4_f32.
// A (16x4 f32) per-lane: lanes 0-15 -> M=lane, K={k,k+1}; lanes 16-31 -> K={k+2,k+3}.
// B (4x16 f32) mirrored: lanes 0-15 -> N=lane, K={k,k+1}; lanes 16-31 -> K={k+2,k+3}.
// C/D: VGPR r -> lanes 0-15 hold (M=r, N=lane); lanes 16-31 hold (M=r+8).
// W staged in LDS pair-interleaved: lW[(kp*Fout + col)*2 + j] = W[2*kp+j][col],
// so the per-lane B fragment {W[kb][col], W[kb+1][col]} is one ds_load_b64.
template <int NT, int K, int Fout>
__global__ void gemm_wmma_f32(const float* __restrict__ X,
                              const float* __restrict__ W,
                              float* __restrict__ Y,
                              int N) {
  static_assert(Fout == NT * 16, "strip must cover Fout");
  __shared__ float lW[K * Fout];      // 64KB (layer1) / 4KB (layer2)

  // cooperative pair-interleaved fill (K, Fout are powers of two -> shifts)
  for (int i = threadIdx.x; i < K * Fout; i += blockDim.x) {
    const int j = i & 1;
    const int rest = i >> 1;
    const int col = rest & (Fout - 1);
    const int kp = rest / Fout;
    lW[i] = W[(size_t)(2 * kp + j) * Fout + col];
  }
  __syncthreads();

  const int lane = threadIdx.x & 31;
  const int wave = threadIdx.x >> 5;
  const int wavesPerBlock = blockDim.x >> 5;
  const int ntilesM = (N + 15) >> 4;
  const int mtile = blockIdx.x * wavesPerBlock + wave;
  if (mtile >= ntilesM) return;       // wave-uniform, after the barrier
  const int mbase = mtile << 4;

  const int m = lane & 15;
  const int khalf = (lane >> 4) << 1; // 0 (lanes 0-15) or 2 (lanes 16-31)
  int row = mbase + m;
  if (row >= N) row = N - 1;          // clamp loads, keep EXEC all-ones
  const float* xrow = X + (size_t)row * K;

  v8f acc[NT];
#pragma unroll
  for (int t = 0; t < NT; ++t) acc[t] = (v8f){};

#pragma unroll 4
  for (int k = 0; k < K; k += 4) {
    const int kb = k + khalf;
    v2f a;
    a.x = xrow[kb];
    a.y = xrow[kb + 1];
    const float* lp = lW + ((size_t)(kb >> 1) * Fout + m) * 2;
#pragma unroll
    for (int t = 0; t < NT; ++t) {
      v2f b = *reinterpret_cast<const v2f*>(lp + t * 32);  // ds_load_b64
      acc[t] = __builtin_amdgcn_wmma_f32_16x16x4_f32(false, a, false, b,
                                                     (short)0, acc[t], false, false);
    }
  }

  const int rbase = mbase + ((lane >> 4) << 3);
#pragma unroll
  for (int t = 0; t < NT; ++t) {
    const int col = t * 16 + m;
#pragma unroll
    for (int r = 0; r < 8; ++r) {
      const int rr = rbase + r;
      if (rr < N) Y[(size_t)rr * Fout + col] = acc[t][r];
    }
  }
}

// ---------------- attention projections: AS/AD[n,h] = <h[n,h,:], att[h,:]> --
__global__ void att_kernel(const float* __restrict__ Hf,
                           const float* __restrict__ att_s,
                           const float* __restrict__ att_d,
                           float* __restrict__ AS, float* __restrict__ AD,
                           int N, int Hh, int C) {
  int i = blockIdx.x * blockDim.x + threadIdx.x;
  if (i >= N * Hh) return;
  int n = i / Hh;
  int h = i - n * Hh;
  const float* hp = Hf + ((size_t)n * Hh + h) * C;
  float s = 0.f, d = 0.f;
  for (int c = 0; c < C; ++c) {
    float v = hp[c];
    s += v * att_s[h * C + c];
    d += v * att_d[h * C + c];
  }
  AS[i] = s;
  AD[i] = d;
}

// ---------------- edge pass 1: segment max of leaky-relu logits ------------
__global__ void edge_max_kernel(const int* __restrict__ src,
                                const int* __restrict__ dst,
                                const float* __restrict__ AS,
                                const float* __restrict__ AD,
                                int* __restrict__ Mk,
                                int EH, int logH) {
  int i = blockIdx.x * blockDim.x + threadIdx.x;
  if (i >= EH) return;
  const int e = i >> logH;
  const int h = i & ((1 << logH) - 1);
  const int s = src[e], d = dst[e];
  const int Hh = 1 << logH;
  float x = lrelu(AS[s * Hh + h] + AD[d * Hh + h]);
  atomicMax(&Mk[d * Hh + h], enc_f32(x));
}

// ---------------- edge pass 2: segment sum of exp(logit - max) -------------
__global__ void edge_sum_kernel(const int* __restrict__ src,
                                const int* __restrict__ dst,
                                const float* __restrict__ AS,
                                const float* __restrict__ AD,
                                const int* __restrict__ Mk,
                                float* __restrict__ S,
                                int EH, int logH) {
  int i = blockIdx.x * blockDim.x + threadIdx.x;
  if (i >= EH) return;
  const int e = i >> logH;
  const int h = i & ((1 << logH) - 1);
  const int s = src[e], d = dst[e];
  const int Hh = 1 << logH;
  const int di = d * Hh + h;
  float x = lrelu(AS[s * Hh + h] + AD[di]);
  float ex = __expf(x - dec_f32(Mk[di]));
  atomicAdd(&S[di], ex);
}

// ---------------- edge pass 3: alpha-weighted message scatter --------------
// C is 8 (layer 1) or 16 (layer 2): gather h[src] with float4 (b128) loads.
__global__ void edge_aggr_kernel(const int* __restrict__ src,
                                 const int* __restrict__ dst,
                                 const float* __restrict__ AS,
                                 const float* __restrict__ AD,
                                 const int* __restrict__ Mk,
                                 const float* __restrict__ S,
                                 const float* __restrict__ Hf,
                                 float* __restrict__ ACC,
                                 int EH, int logH, int C4) {  // C4 = C/4
  int i = blockIdx.x * blockDim.x + threadIdx.x;
  if (i >= EH) return;
  const int e = i >> logH;
  const int h = i & ((1 << logH) - 1);
  const int s = src[e], d = dst[e];
  const int Hh = 1 << logH;
  const int di = d * Hh + h;
  float x = lrelu(AS[s * Hh + h] + AD[di]);
  float alpha = __expf(x - dec_f32(Mk[di])) / (S[di] + 1e-16f);
  const float4* hp = (const float4*)(Hf + ((size_t)s * Hh + h) * (C4 * 4));
  float* op = ACC + ((size_t)d * Hh + h) * (C4 * 4);
  for (int q = 0; q < C4; ++q) {
    float4 v = hp[q];
    atomicAdd(&op[q * 4 + 0], v.x * alpha);
    atomicAdd(&op[q * 4 + 1], v.y * alpha);
    atomicAdd(&op[q * 4 + 2], v.z * alpha);
    atomicAdd(&op[q * 4 + 3], v.w * alpha);
  }
}

// ---------------- bias + ELU (in place, layer-1 output -> layer-2 input) ---
__global__ void bias_elu_kernel(float* __restrict__ A,
                                const float* __restrict__ b,
                                int N, int F) {
  int i = blockIdx.x * blockDim.x + threadIdx.x;
  if (i >= N * F) return;
  float v = A[i] + b[i % F];
  A[i] = v > 0.0f ? v : (__expf(v) - 1.0f);
}

// ---------------- final: out = log_softmax(ACC2 + b2) over 16 classes ------
__global__ void final_logsoftmax_kernel(const float* __restrict__ ACC,
                                        const float* __restrict__ b,
                                        float* __restrict__ out, int N) {
  int n = blockIdx.x * blockDim.x + threadIdx.x;
  if (n >= N) return;
  float vals[16];
  float mx = -3.402823466e38f;
  for (int c = 0; c < 16; ++c) {
    float v = ACC[(size_t)n * 16 + c] + b[c];
    vals[c] = v;
    mx = v > mx ? v : mx;
  }
  float se = 0.f;
  for (int c = 0; c < 16; ++c) se += __expf(vals[c] - mx);
  float lse = mx + __logf(se);
  for (int c = 0; c < 16; ++c) out[(size_t)n * 16 + c] = vals[c] - lse;
}

// ---------------------------------------------------------------------------
extern "C" void kernel_launch(void* const* d_in, const int* in_sizes, int n_in,
                              void* d_out, int out_size, void* d_ws, size_t ws_size,
                              hipStream_t stream) {
  const float* x        = (const float*)d_in[0];
  const int*   ei       = (const int*)d_in[1];   // [2, E] (src row, dst row)
  const float* W1       = (const float*)d_in[2]; // [256, 64]
  const float* att_src1 = (const float*)d_in[3]; // [8, 8]
  const float* att_dst1 = (const float*)d_in[4];
  const float* b1       = (const float*)d_in[5]; // [64]
  const float* W2       = (const float*)d_in[6]; // [64, 16]
  const float* att_src2 = (const float*)d_in[7]; // [1, 16]
  const float* att_dst2 = (const float*)d_in[8];
  const float* b2       = (const float*)d_in[9]; // [16]
  float* out = (float*)d_out;

  const int F_IN = 256, F1 = 64, H1 = 8, C1 = 8, NC = 16;
  const int N = in_sizes[0] / F_IN;   // 100000
  const int E = in_sizes[1] / 2;      // 1600000
  const int* srcI = ei;
  const int* dstI = ei + E;

  // workspace layout (floats): ~196 floats/node ≈ 78.4 MB total
  float* H1f  = (float*)d_ws;          // [N, 64]
  float* AS1  = H1f + (size_t)N * F1;  // [N, 8]
  float* AD1  = AS1 + (size_t)N * H1;  // [N, 8]
  int*   M1   = (int*)(AD1 + (size_t)N * H1);   // [N, 8] encoded max keys
  float* S1   = (float*)(M1 + (size_t)N * H1);  // [N, 8]
  float* ACC1 = S1 + (size_t)N * H1;   // [N, 64] aggregate -> ELU in place
  float* H2f  = ACC1 + (size_t)N * F1; // [N, 16]
  float* AS2  = H2f + (size_t)N * NC;  // [N]
  float* AD2  = AS2 + N;               // [N]
  int*   M2   = (int*)(AD2 + N);       // [N]
  float* S2   = (float*)(M2 + N);      // [N]
  float* ACC2 = S2 + N;                // [N, 16]

  const int TB = 256;
  auto blocks = [](long long n, int tb) { return (int)((n + tb - 1) / tb); };

  // --- re-init accumulators / max keys (ws is poisoned by harness) ---------
  fill_i32<<<blocks((long long)N * H1, TB), TB, 0, stream>>>(M1, (int)0x80000000, N * H1);
  fill_i32<<<blocks(N, TB), TB, 0, stream>>>(M2, (int)0x80000000, N);
  fill_f32<<<blocks((long long)N * H1, TB), TB, 0, stream>>>(S1, 0.f, N * H1);
  fill_f32<<<blocks((long long)N * F1, TB), TB, 0, stream>>>(ACC1, 0.f, N * F1);
  fill_f32<<<blocks(N, TB), TB, 0, stream>>>(S2, 0.f, N);
  fill_f32<<<blocks((long long)N * NC, TB), TB, 0, stream>>>(ACC2, 0.f, N * NC);

  // --- layer 1 -------------------------------------------------------------
  {
    const int ntilesM = (N + 15) / 16;   // one wave per 16-row strip; 8 waves/block
    gemm_wmma_f32<4, 256, 64><<<blocks(ntilesM, 8), 256, 0, stream>>>(x, W1, H1f, N);
  }
  att_kernel<<<blocks((long long)N * H1, TB), TB, 0, stream>>>(
      H1f, att_src1, att_dst1, AS1, AD1, N, H1, C1);

  const int EH1 = E * H1;  // 12.8M
  edge_max_kernel<<<blocks(EH1, TB), TB, 0, stream>>>(srcI, dstI, AS1, AD1, M1, EH1, 3);
  edge_sum_kernel<<<blocks(EH1, TB), TB, 0, stream>>>(srcI, dstI, AS1, AD1, M1, S1, EH1, 3);
  edge_aggr_kernel<<<blocks(EH1, TB), TB, 0, stream>>>(srcI, dstI, AS1, AD1, M1, S1,
                                                       H1f, ACC1, EH1, 3, C1 / 4);
  bias_elu_kernel<<<blocks((long long)N * F1, TB), TB, 0, stream>>>(ACC1, b1, N, F1);

  // --- layer 2 -------------------------------------------------------------
  {
    const int ntilesM = (N + 15) / 16;
    gemm_wmma_f32<1, 64, 16><<<blocks(ntilesM, 8), 256, 0, stream>>>(ACC1, W2, H2f, N);
  }
  att_kernel<<<blocks(N, TB), TB, 0, stream>>>(H2f, att_src2, att_dst2, AS2, AD2, N, 1, NC);

  edge_max_kernel<<<blocks(E, TB), TB, 0, stream>>>(srcI, dstI, AS2, AD2, M2, E, 0);
  edge_sum_kernel<<<blocks(E, TB), TB, 0, stream>>>(srcI, dstI, AS2, AD2, M2, S2, E, 0);
  edge_aggr_kernel<<<blocks(E, TB), TB, 0, stream>>>(srcI, dstI, AS2, AD2, M2, S2,
                                                     H2f, ACC2, E, 0, NC / 4);

  // --- final bias + log_softmax -------------------------------------------
  final_logsoftmax_kernel<<<blocks(N, TB), TB, 0, stream>>>(ACC2, b2, out, N);
}